// FC_AMMD_66400194396210
// MI455X (gfx1250) — compile-verified
//
#include <hip/hip_runtime.h>
#include <hip/hip_bf16.h>
#include <math.h>

// ---------------- constants matching the reference ----------------
#define NWAY   5
#define MAXC   256      // MAX_PER_CLASS
#define NS_TOT 16384    // support samples per class row
#define NQ_TOT 65536    // query samples per chunk
#define DIM    256      // feature dim
#define GAMMA  0.1f
#define LAMBDA 0.1f

typedef __attribute__((ext_vector_type(2))) float v2f;
typedef __attribute__((ext_vector_type(8))) float v8f;

// =================================================================
// 1) Ordered wave-ballot compaction of support indices (first 256).
// =================================================================
__global__ void k_gather_sup(const int* __restrict__ l_s,
                             int* __restrict__ idxS, int* __restrict__ nsFull,
                             int* __restrict__ naSel) {
  int c = blockIdx.x;
  int lane = threadIdx.x;            // blockDim == 32 (one wave, wave32)
  const int* lab = l_s + c * NS_TOT;
  int cnt = 0;
  for (int base = 0; base < NS_TOT; base += 32) {
    int pred = (lab[base + lane] == 1);
    unsigned long long m = __ballot(pred);
    int ofs = __popcll(m & ((1ull << lane) - 1ull));
    if (pred && (cnt + ofs) < MAXC) idxS[c * MAXC + cnt + ofs] = base + lane;
    cnt += __popcll(m);
  }
  if (lane == 0) { nsFull[c] = cnt; naSel[c] = cnt < MAXC ? cnt : MAXC; }
}

__global__ void k_gather_qry(const int* __restrict__ l_q,
                             int* __restrict__ idxQ, int* __restrict__ nqFull,
                             int* __restrict__ nbSel) {
  int c = blockIdx.x;
  int lane = threadIdx.x;            // blockDim == 32
  int cnt = 0;
  const int TOT = NWAY * NQ_TOT;     // row-major flat scan == np.nonzero order
  for (int base = 0; base < TOT; base += 32) {
    int pred = (l_q[base + lane] == (c + 1));
    unsigned long long m = __ballot(pred);
    int ofs = __popcll(m & ((1ull << lane) - 1ull));
    if (pred && (cnt + ofs) < MAXC) idxQ[c * MAXC + cnt + ofs] = base + lane;
    cnt += __popcll(m);
  }
  if (lane == 0) { nqFull[c] = cnt; nbSel[c] = cnt < MAXC ? cnt : MAXC; }
}

// =================================================================
// 2) Gather + row-normalize into dense padded XA/XB (256x256/class).
// =================================================================
__global__ void k_build_X(const float* __restrict__ f_s, const float* __restrict__ f_q,
                          const int* __restrict__ idxS, const int* __restrict__ idxQ,
                          const int* __restrict__ naSel, const int* __restrict__ nbSel,
                          float* __restrict__ XA, float* __restrict__ XB,
                          float* __restrict__ x2A, float* __restrict__ x2B) {
  int r   = blockIdx.x;   // row 0..255
  int c   = blockIdx.y;   // class
  int isB = blockIdx.z;   // 0 = support, 1 = query
  int d   = threadIdx.x;  // dim 0..255
  __shared__ float red[DIM];
  float* X  = (isB ? XB : XA) + ((size_t)c * MAXC + r) * DIM;
  float* x2 = (isB ? x2B : x2A) + c * MAXC + r;
  int n = isB ? nbSel[c] : naSel[c];
  float val = 0.f;
  if (r < n) {
    if (!isB) {
      int si = idxS[c * MAXC + r];
      val = f_s[((size_t)c * DIM + d) * NS_TOT + si];
    } else {
      int qi = idxQ[c * MAXC + r];
      int m = qi >> 16, nn = qi & (NQ_TOT - 1);
      val = f_q[((size_t)m * DIM + d) * NQ_TOT + nn];
    }
  }
  red[d] = val * val;
  __syncthreads();
  for (int s = DIM / 2; s > 0; s >>= 1) { if (d < s) red[d] += red[d + s]; __syncthreads(); }
  float ss  = red[0];
  float inv = 1.f / (sqrtf(ss) + 1e-12f);
  X[d] = (r < n) ? val * inv : 0.f;
  if (d == 0) *x2 = (r < n) ? ss * inv * inv : 0.f;
}

// =================================================================
// 3) 35 pairwise-sq-distance matrices via V_WMMA_F32_16X16X4_F32.
//    Block (4 waves) owns a 16-row x 256-col output slab of one job.
//    The 16x256 A row-slab (16KB) is staged to LDS with the CDNA5
//    async-to-LDS path (ASYNCcnt), then each wave runs 4 independent
//    accumulator chains (tn tiles), sharing one A fragment per K-step.
// =================================================================
__global__ void k_pairwise_wmma(const float* __restrict__ XA, const float* __restrict__ XB,
                                const float* __restrict__ x2A, const float* __restrict__ x2B,
                                float* __restrict__ Daa, float* __restrict__ Dbb,
                                float* __restrict__ Dab) {
  int j = blockIdx.y;
  const float *A, *B, *x2a, *x2b; float* D;
  if (j < 5)       { A = XA + (size_t)j * MAXC * DIM; B = A;
                     x2a = x2A + j * MAXC; x2b = x2a; D = Daa + (size_t)j * MAXC * MAXC; }
  else if (j < 10) { int c = j - 5; A = XB + (size_t)c * MAXC * DIM; B = A;
                     x2a = x2B + c * MAXC; x2b = x2a; D = Dbb + (size_t)c * MAXC * MAXC; }
  else             { int k = (j - 10) / 5, c = (j - 10) % 5;
                     A = XA + (size_t)k * MAXC * DIM; B = XB + (size_t)c * MAXC * DIM;
                     x2a = x2A + k * MAXC; x2b = x2B + c * MAXC;
                     D = Dab + (size_t)(j - 10) * MAXC * MAXC; }

  __shared__ float shA[16 * DIM];               // 16KB A row-slab
  int tm   = blockIdx.x;                        // 16 row-slabs
  int wave = threadIdx.x >> 5;                  // 4 waves / block
  int lane = threadIdx.x & 31;
  int half = lane >> 4, l16 = lane & 15;

  // ---- async stage: contiguous 16KB slab A[tm*16 .. tm*16+15][:] -> LDS ----
  {
    const float* Aslab = A + (size_t)tm * 16 * DIM;
    unsigned lbase = (unsigned)(size_t)(&shA[0]);
    #pragma unroll
    for (int it = 0; it < 8; ++it) {            // 128 thr * 16B * 8 = 16KB
      int e = it * 128 + threadIdx.x;
      unsigned long long ga = (unsigned long long)(size_t)Aslab
                            + (unsigned long long)e * 16ull;
      unsigned la = lbase + (unsigned)e * 16u;
      asm volatile("global_load_async_to_lds_b128 %0, %1, off"
                   :: "v"(la), "v"(ga) : "memory");
    }
    asm volatile("s_wait_asynccnt 0" ::: "memory");
    __syncthreads();                            // other waves' stages too
  }

  // ---- 4 independent WMMA chains per wave (tn = wave*4 + t) ----
  const float* Bp0 = B + (size_t)((wave * 4 + 0) * 16 + l16) * DIM + 2 * half;
  const float* Bp1 = B + (size_t)((wave * 4 + 1) * 16 + l16) * DIM + 2 * half;
  const float* Bp2 = B + (size_t)((wave * 4 + 2) * 16 + l16) * DIM + 2 * half;
  const float* Bp3 = B + (size_t)((wave * 4 + 3) * 16 + l16) * DIM + 2 * half;
  const float* Ap  = &shA[l16 * DIM + 2 * half];

  v8f acc0 = {}, acc1 = {}, acc2 = {}, acc3 = {};
  #pragma unroll 4
  for (int k0 = 0; k0 < DIM; k0 += 4) {
    v2f a  = *(const v2f*)(Ap  + k0);           // ds_load_b64 (shared A frag)
    v2f b0 = *(const v2f*)(Bp0 + k0);
    v2f b1 = *(const v2f*)(Bp1 + k0);
    v2f b2 = *(const v2f*)(Bp2 + k0);
    v2f b3 = *(const v2f*)(Bp3 + k0);
    acc0 = __builtin_amdgcn_wmma_f32_16x16x4_f32(false, a, false, b0, (short)0, acc0, false, false);
    acc1 = __builtin_amdgcn_wmma_f32_16x16x4_f32(false, a, false, b1, (short)0, acc1, false, false);
    acc2 = __builtin_amdgcn_wmma_f32_16x16x4_f32(false, a, false, b2, (short)0, acc2, false, false);
    acc3 = __builtin_amdgcn_wmma_f32_16x16x4_f32(false, a, false, b3, (short)0, acc3, false, false);
  }

  // ---- epilogue: D = max(x2a_i + x2b_j - 2*G, 0) ----
  float xr[8];
  #pragma unroll
  for (int v = 0; v < 8; v++) xr[v] = x2a[tm * 16 + v + 8 * half];
  #pragma unroll
  for (int t = 0; t < 4; t++) {
    v8f acc = (t == 0) ? acc0 : (t == 1) ? acc1 : (t == 2) ? acc2 : acc3;
    int col = (wave * 4 + t) * 16 + l16;
    float xc = x2b[col];
    #pragma unroll
    for (int v = 0; v < 8; v++) {
      int row = tm * 16 + v + 8 * half;         // C/D layout: lane>=16 -> M += 8
      D[(size_t)row * MAXC + col] = fmaxf(xr[v] + xc - 2.0f * acc[v], 0.0f);
    }
  }
}

// =================================================================
// 4) Exact median of the concatenated-z distance multiset
//    {Daa_k x1, Dab_kc x2, Dbb_c x1} by 3-pass radix select on the
//    float bit pattern (11+11+9 bits) -- 4 scans instead of 31.
// =================================================================
__device__ __forceinline__ void
histScan(const float* Pa, const float* Pab, const float* Pb, int n1, int n2,
         unsigned prefix, int hs, int sh, unsigned mask, int* hist, int tid) {
  for (int r = 0; r < n1; r++) { const float* row = Pa + r * MAXC;
    for (int q = tid; q < n1; q += 256) {
      unsigned u = __float_as_uint(row[q]);
      if ((u >> hs) == prefix) atomicAdd(&hist[(u >> sh) & mask], 1); } }
  for (int r = 0; r < n1; r++) { const float* row = Pab + r * MAXC;
    for (int q = tid; q < n2; q += 256) {
      unsigned u = __float_as_uint(row[q]);
      if ((u >> hs) == prefix) atomicAdd(&hist[(u >> sh) & mask], 2); } }
  for (int r = 0; r < n2; r++) { const float* row = Pb + r * MAXC;
    for (int q = tid; q < n2; q += 256) {
      unsigned u = __float_as_uint(row[q]);
      if ((u >> hs) == prefix) atomicAdd(&hist[(u >> sh) & mask], 1); } }
}

__global__ void k_median(const float* __restrict__ Daa, const float* __restrict__ Dbb,
                         const float* __restrict__ Dab,
                         const int* __restrict__ naSel, const int* __restrict__ nbSel,
                         const int* __restrict__ nsFull, const int* __restrict__ nqFull,
                         float* __restrict__ sigbase) {
  int p = blockIdx.x;            // 25 pairs, p = k*5 + c
  int k = p / 5, c = p % 5;
  int tid = threadIdx.x;         // 256 threads
  __shared__ int hist[2048];
  __shared__ int sh_nz, sh_base;
  __shared__ unsigned sh_sel;
  if (!((nsFull[k] >= 2) && (nqFull[c] >= 2))) { if (tid == 0) sigbase[p] = 1.0f; return; }
  int n1 = naSel[k], n2 = nbSel[c];
  const float* Pa  = Daa + (size_t)k * MAXC * MAXC;
  const float* Pb  = Dbb + (size_t)c * MAXC * MAXC;
  const float* Pab = Dab + (size_t)p * MAXC * MAXC;
  int T = n1 * n1 + 2 * n1 * n2 + n2 * n2;     // <= 262144

  // nz = #(d <= 0)  (distances are clamped non-negative)
  int nzloc = 0;
  for (int r = 0; r < n1; r++) { const float* row = Pa + r * MAXC;
    for (int q = tid; q < n1; q += 256) nzloc += (row[q] <= 0.0f); }
  for (int r = 0; r < n1; r++) { const float* row = Pab + r * MAXC;
    for (int q = tid; q < n2; q += 256) nzloc += 2 * (row[q] <= 0.0f); }
  for (int r = 0; r < n2; r++) { const float* row = Pb + r * MAXC;
    for (int q = tid; q < n2; q += 256) nzloc += (row[q] <= 0.0f); }
  if (tid == 0) sh_nz = 0;
  __syncthreads();
  atomicAdd(&sh_nz, nzloc);
  __syncthreads();
  int nz = sh_nz;

  float base;
  if (nz >= T) base = 1.0f;
  else {
    int target = nz + (T - nz - 1) / 2;        // 0-indexed order statistic
    if (target > T - 1) target = T - 1;
    unsigned prefix = 0; int below = 0;
    const int shifts[3] = {20, 9, 0};
    const int nbits [3] = {11, 11, 9};
    for (int pass = 0; pass < 3; ++pass) {
      int sh = shifts[pass], nb = nbits[pass], hs = sh + nb;
      unsigned mask = (1u << nb) - 1u;
      int nbk = 1 << nb;
      for (int b = tid; b < nbk; b += 256) hist[b] = 0;
      __syncthreads();
      histScan(Pa, Pab, Pb, n1, n2, prefix, hs, sh, mask, hist, tid);
      __syncthreads();
      if (tid == 0) {
        int cum = below; unsigned sel = 0;
        for (int b = 0; b < nbk; b++) {
          if (cum + hist[b] > target) { sel = (unsigned)b; break; }
          cum += hist[b];
        }
        sh_sel = sel; sh_base = cum;
      }
      __syncthreads();
      prefix = (prefix << nb) | sh_sel;
      below  = sh_base;
    }
    base = sqrtf(__uint_as_float(prefix) + 1e-6f);  // exact k-th smallest
  }
  if (tid == 0) sigbase[p] = base;
}

// =================================================================
// 5) Multi-sigma RBF means + MMD^2 for all 25 (k,c) pairs.
// =================================================================
__device__ __forceinline__ float
blockReduceF(float v, float* red, int tid) {
  red[tid] = v; __syncthreads();
  for (int s = 128; s > 0; s >>= 1) { if (tid < s) red[tid] += red[tid + s]; __syncthreads(); }
  float r = red[0]; __syncthreads();
  return r;
}

__device__ __forceinline__ float
sumRBF(const float* D, int nr, int nc, const float* inv, int tid, float* red) {
  float acc = 0.f;
  for (int r = 0; r < nr; r++) {
    const float* row = D + r * MAXC;
    for (int q = tid; q < nc; q += 256) {
      float d = row[q];
      acc += __expf(-d * inv[0]) + __expf(-d * inv[1])
           + __expf(-d * inv[2]) + __expf(-d * inv[3]);
    }
  }
  return blockReduceF(acc, red, tid);
}

__global__ void k_mmd(const float* __restrict__ Daa, const float* __restrict__ Dbb,
                      const float* __restrict__ Dab,
                      const int* __restrict__ naSel, const int* __restrict__ nbSel,
                      const int* __restrict__ nsFull, const int* __restrict__ nqFull,
                      const float* __restrict__ sigbase, float* __restrict__ mmd) {
  int p = blockIdx.x; int k = p / 5, c = p % 5; int tid = threadIdx.x;
  __shared__ float red[256];
  if (!((nsFull[k] >= 2) && (nqFull[c] >= 2))) { if (tid == 0) mmd[p] = 0.f; return; }
  int n1 = naSel[k], n2 = nbSel[c];
  float b = sigbase[p];
  const float grid[4] = {0.5f, 1.0f, 2.0f, 4.0f};
  float inv[4];
  #pragma unroll
  for (int s = 0; s < 4; s++) { float sg = b * grid[s]; inv[s] = 1.f / (2.f * sg * sg); }
  float sa  = sumRBF(Daa + (size_t)k * MAXC * MAXC, n1, n1, inv, tid, red);
  float sb  = sumRBF(Dbb + (size_t)c * MAXC * MAXC, n2, n2, inv, tid, red);
  float sab = sumRBF(Dab + (size_t)p * MAXC * MAXC, n1, n2, inv, tid, red);
  if (tid == 0) {
    float m = sa / ((float)n1 * (float)n1) + sb / ((float)n2 * (float)n2)
            - 2.f * sab / ((float)n1 * (float)n2);
    mmd[p] = fmaxf(m, 0.f);
  }
}

// =================================================================
// 6) Final scalar combine (HARD_POOL=8 >= 4 negs => pool = all negs;
//    NUM_NEG=1 => pick = argmin over per-pair-sigma MMD^2).
// =================================================================
__global__ void k_final(const float* __restrict__ mmd,
                        const int* __restrict__ nsFull, const int* __restrict__ nqFull,
                        float* __restrict__ out) {
  float total = 0.f, vw = 0.f;
  for (int c = 0; c < NWAY; c++) {
    if (!((nsFull[c] >= 2) && (nqFull[c] >= 2))) continue;
    float pos = mmd[c * 5 + c];
    float best = 3.402823466e+38f; int any = 0;
    for (int k = 0; k < NWAY; k++) {
      if (k == c || nsFull[k] < 2) continue;
      float v = mmd[k * 5 + c];
      if (v < best) { best = v; }
      any = 1;
    }
    float loss_c = any ? fmaxf(pos - best + GAMMA, 0.f) : pos;
    float w = sqrtf((float)nsFull[c] * (float)nqFull[c]);
    total += w * loss_c; vw += w;
  }
  out[0] = LAMBDA * total / fmaxf(vw, 1e-12f);
}

// =================================================================
// host-side launcher
// =================================================================
extern "C" void kernel_launch(void* const* d_in, const int* in_sizes, int n_in,
                              void* d_out, int out_size, void* d_ws, size_t ws_size,
                              hipStream_t stream) {
  const float* f_s = (const float*)d_in[0];
  const int*   l_s = (const int*)  d_in[1];
  const float* f_q = (const float*)d_in[2];
  const int*   l_q = (const int*)  d_in[3];
  float* out = (float*)d_out;

  // workspace carving (all offsets 256B aligned); total ~11.8 MB
  char* w = (char*)d_ws;
  auto take = [&](size_t bytes) { char* p = w; w += (bytes + 255) & ~((size_t)255); return p; };
  int*   idxS    = (int*)  take(NWAY * MAXC * sizeof(int));
  int*   idxQ    = (int*)  take(NWAY * MAXC * sizeof(int));
  int*   nsFull  = (int*)  take(NWAY * sizeof(int));
  int*   naSel   = (int*)  take(NWAY * sizeof(int));
  int*   nqFull  = (int*)  take(NWAY * sizeof(int));
  int*   nbSel   = (int*)  take(NWAY * sizeof(int));
  float* x2A     = (float*)take(NWAY * MAXC * sizeof(float));
  float* x2B     = (float*)take(NWAY * MAXC * sizeof(float));
  float* XA      = (float*)take((size_t)NWAY * MAXC * DIM * sizeof(float));
  float* XB      = (float*)take((size_t)NWAY * MAXC * DIM * sizeof(float));
  float* Daa     = (float*)take((size_t)NWAY * MAXC * MAXC * sizeof(float));
  float* Dbb     = (float*)take((size_t)NWAY * MAXC * MAXC * sizeof(float));
  float* Dab     = (float*)take((size_t)NWAY * NWAY * MAXC * MAXC * sizeof(float));
  float* sigbase = (float*)take(NWAY * NWAY * sizeof(float));
  float* mmd     = (float*)take(NWAY * NWAY * sizeof(float));

  k_gather_sup<<<NWAY, 32, 0, stream>>>(l_s, idxS, nsFull, naSel);
  k_gather_qry<<<NWAY, 32, 0, stream>>>(l_q, idxQ, nqFull, nbSel);
  k_build_X<<<dim3(MAXC, NWAY, 2), DIM, 0, stream>>>(f_s, f_q, idxS, idxQ,
                                                     naSel, nbSel, XA, XB, x2A, x2B);
  k_pairwise_wmma<<<dim3(16, 35), 128, 0, stream>>>(XA, XB, x2A, x2B, Daa, Dbb, Dab);
  k_median<<<NWAY * NWAY, 256, 0, stream>>>(Daa, Dbb, Dab, naSel, nbSel,
                                            nsFull, nqFull, sigbase);
  k_mmd<<<NWAY * NWAY, 256, 0, stream>>>(Daa, Dbb, Dab, naSel, nbSel,
                                         nsFull, nqFull, sigbase, mmd);
  k_final<<<1, 1, 0, stream>>>(mmd, nsFull, nqFull, out);
}